// softConLoss_55860344652215
// MI455X (gfx1250) — compile-verified
//
#include <hip/hip_runtime.h>
#include <math.h>

// ---------------- problem constants ----------------
#define NROW 8192
#define DDIM 128
#define NCLS 10
#define INV_TEMP 10.0f     // 1 / 0.1
#define BM 128             // rows per block (8 waves x 16)
#define BN 64              // j rows staged per iteration
#define JS 8               // j-range splits (grid.y)
#define LDB 132            // padded LDS row stride (floats) -> conflict-free b64 gathers

typedef __attribute__((ext_vector_type(2))) float v2f;
typedef __attribute__((ext_vector_type(8))) float v8f;
typedef __attribute__((ext_vector_type(4))) int   v4i;

// async global->LDS b128 copy (gfx1250 ASYNCcnt path)
__device__ __forceinline__ void async_b128(const float* g, float* l) {
  __builtin_amdgcn_global_load_async_to_lds_b128(
      (__attribute__((address_space(1))) v4i*)g,
      (__attribute__((address_space(3))) v4i*)l, 0, 0);
}

// ---------------- kernel 0: zero scratch + output ----------------
__global__ void zero_kernel(float* __restrict__ s, float* __restrict__ cnt,
                            float* __restrict__ out) {
  int i = blockIdx.x * blockDim.x + threadIdx.x;
  if (i < NROW * NCLS) s[i] = 0.0f;
  if (i < NCLS) cnt[i] = 0.0f;
  if (i == 0) out[0] = 0.0f;
}

// ---------------- kernel 1: class ids + class counts ----------------
__global__ void prep_kernel(const float* __restrict__ label,
                            int* __restrict__ cls, float* __restrict__ cnt) {
  int i = blockIdx.x * blockDim.x + threadIdx.x;
  if (i >= NROW) return;
  const float* L = label + (size_t)i * NCLS;
  int cb = 0;
  float vb = L[0];
#pragma unroll
  for (int c = 1; c < NCLS; ++c) {
    float v = L[c];
    if (v > vb) { vb = v; cb = c; }
  }
  cls[i] = cb;
  unsafeAtomicAdd(&cnt[cb], 1.0f);
}

// ---------------- kernel 2: fused exp-GEMM + class reduction ----------------
// s[i,c] = sum_{j != i} exp((f_i . f_j) * INV_TEMP) * [cls_j == c]
__global__ __launch_bounds__(256) void sim_kernel(const float* __restrict__ feat,
                                                  const int* __restrict__ cls,
                                                  float* __restrict__ s) {
  __shared__ float Bt[2 * BN * LDB];      // double-buffered staged j-rows
  __shared__ float sacc[BM * NCLS];       // per-block class accumulators

  const int tid  = threadIdx.x;
  const int lane = tid & 31;
  const int wave = tid >> 5;
  const int i0   = blockIdx.x * BM;
  const int jbase = blockIdx.y * (NROW / JS);
  const int NT = (NROW / JS) / BN;

  for (int t = tid; t < BM * NCLS; t += 256) sacc[t] = 0.0f;

  // A fragments: lane m=lane&15 holds row (i0 + wave*16 + m),
  // K pair (kc*4 + (lane>>4)*2) per ISA 32-bit A layout.
  const int m  = lane & 15;
  const int kh = lane >> 4;
  const float* abase = feat + (size_t)(i0 + wave * 16 + m) * DDIM + kh * 2;
  v2f a[32];
#pragma unroll
  for (int kc = 0; kc < 32; ++kc) a[kc] = *(const v2f*)(abase + kc * 4);

  // prologue: async-stage first tile into buffer 0
#pragma unroll
  for (int c = 0; c < (BN * DDIM) / (256 * 4); ++c) {
    int f = c * 1024 + tid * 4;
    int row = f >> 7, col = f & 127;
    async_b128(feat + (size_t)jbase * DDIM + f, &Bt[row * LDB + col]);
  }

#pragma unroll 1
  for (int jt = 0; jt < NT; ++jt) {
    const int cur = jt & 1;
    const int j0 = jbase + jt * BN;

    // staged buffer complete + all waves done with the buffer we stage next
    __builtin_amdgcn_s_wait_asynccnt(0);
    __syncthreads();

    // async-stage next tile into the other buffer while we compute
    if (jt + 1 < NT) {
      float* nb = &Bt[(cur ^ 1) * (BN * LDB)];
      const float* ng = feat + (size_t)(j0 + BN) * DDIM;
#pragma unroll
      for (int c = 0; c < (BN * DDIM) / (256 * 4); ++c) {
        int f = c * 1024 + tid * 4;
        int row = f >> 7, col = f & 127;
        async_b128(ng + f, &nb[row * LDB + col]);
      }
    }

    const float* Bc = &Bt[cur * (BN * LDB)];
#pragma unroll 1
    for (int js = 0; js < BN / 16; ++js) {
      const int nloc = js * 16 + m;          // local j row; lane&15 doubles as N
      const int j = j0 + nloc;
      const float* bbase = &Bc[nloc * LDB + kh * 2];

      v8f acc0 = {0,0,0,0,0,0,0,0}, acc1 = {0,0,0,0,0,0,0,0};
      v8f acc2 = {0,0,0,0,0,0,0,0}, acc3 = {0,0,0,0,0,0,0,0};
#pragma unroll
      for (int kc = 0; kc < 32; kc += 4) {
        v2f b0 = *(const v2f*)(bbase + (kc + 0) * 4);
        v2f b1 = *(const v2f*)(bbase + (kc + 1) * 4);
        v2f b2 = *(const v2f*)(bbase + (kc + 2) * 4);
        v2f b3 = *(const v2f*)(bbase + (kc + 3) * 4);
        acc0 = __builtin_amdgcn_wmma_f32_16x16x4_f32(false, a[kc + 0], false, b0,
                                                     (short)0, acc0, false, false);
        acc1 = __builtin_amdgcn_wmma_f32_16x16x4_f32(false, a[kc + 1], false, b1,
                                                     (short)0, acc1, false, false);
        acc2 = __builtin_amdgcn_wmma_f32_16x16x4_f32(false, a[kc + 2], false, b2,
                                                     (short)0, acc2, false, false);
        acc3 = __builtin_amdgcn_wmma_f32_16x16x4_f32(false, a[kc + 3], false, b3,
                                                     (short)0, acc3, false, false);
      }
      v8f acc = (acc0 + acc1) + (acc2 + acc3);

      // epilogue: diag mask, exp, class-bucketed LDS accumulation
      const int cj = cls[j];
      const int rbase = wave * 16 + 8 * kh;  // D layout: VGPR r -> M = r + 8*(lane>>4)
#pragma unroll
      for (int r = 0; r < 8; ++r) {
        const int gi = i0 + rbase + r;
        float t = (gi == j) ? 0.0f : __expf(acc[r] * INV_TEMP);
        atomicAdd(&sacc[(rbase + r) * NCLS + cj], t);
      }
    }
  }

  __syncthreads();
  for (int t = tid; t < BM * NCLS; t += 256)
    unsafeAtomicAdd(&s[(size_t)i0 * NCLS + t], sacc[t]);
}

// ---------------- kernel 3: per-row log-softmax loss + mean ----------------
__global__ void loss_kernel(const float* __restrict__ s,
                            const float* __restrict__ cnt,
                            const int* __restrict__ cls,
                            float* __restrict__ out) {
  int i = blockIdx.x * blockDim.x + threadIdx.x;
  if (i < NCLS || i >= NROW) return;  // rows [NUM_CLASS:] only
  const int ci = cls[i];
  float l[NCLS];
  float mx = -1e30f;
#pragma unroll
  for (int c = 0; c < NCLS; ++c) {
    float py = cnt[c] - ((c == ci) ? 1.0f : 0.0f);
    float v = logf(s[(size_t)i * NCLS + c]) - logf(py);
    l[c] = v;
    mx = fmaxf(mx, v);
  }
  float sum = 0.0f;
#pragma unroll
  for (int c = 0; c < NCLS; ++c) sum += __expf(l[c] - mx);
  float lse = mx + logf(sum);
  float loss_i = lse - l[ci];
  unsafeAtomicAdd(out, loss_i * (1.0f / (float)(NROW - NCLS)));
}

// ---------------- host-side launcher ----------------
extern "C" void kernel_launch(void* const* d_in, const int* in_sizes, int n_in,
                              void* d_out, int out_size, void* d_ws, size_t ws_size,
                              hipStream_t stream) {
  const float* label = (const float*)d_in[0];  // (8192, 10) fp32 one-hot
  const float* feat  = (const float*)d_in[1];  // (8192, 128) fp32
  float* out = (float*)d_out;                  // scalar loss

  float* wsf = (float*)d_ws;
  float* s   = wsf;                            // 8192*10 floats
  float* cnt = wsf + NROW * NCLS;              // 10 floats
  int*   cls = (int*)(wsf + NROW * NCLS + 16); // 8192 ints

  zero_kernel<<<(NROW * NCLS + 255) / 256, 256, 0, stream>>>(s, cnt, out);
  prep_kernel<<<NROW / 256, 256, 0, stream>>>(label, cls, cnt);
  dim3 grid(NROW / BM, JS);
  sim_kernel<<<grid, 256, 0, stream>>>(feat, cls, s);
  loss_kernel<<<NROW / 256, 256, 0, stream>>>(s, cnt, cls, out);
}